// RoPE_61873298866730
// MI455X (gfx1250) — compile-verified
//
#include <hip/hip_runtime.h>
#include <cstdint>

// RoPE, fp32, x:(4,16,4096,128). Pure bandwidth problem (~256 MiB HBM traffic,
// floor ~11us @ 23.3 TB/s). Strategy:
//  - 2D grid: x-dim = sequence blocks (32 rows each), y-dim = B*H  -> no
//    integer division/modulo anywhere in the kernel.
//  - stage 32 sin rows + 32 cos rows (16 KB) into LDS with gfx1250 async
//    global->LDS b128 copies (ASYNCcnt path), s_wait_asynccnt + barrier.
//  - stream x with non-temporal b128 loads/stores, rotate pairs with FMAs,
//    sin/cos served from LDS (each staged value read once per block).

typedef float v4f __attribute__((ext_vector_type(4)));
typedef float v2f __attribute__((ext_vector_type(2)));

#define ROWS_PER_BLOCK 32   // s-rows per block (S=4096 % 32 == 0)
#define HALF_D 64           // D_K/2
#define THREADS 256         // 8 wave32s

__global__ __launch_bounds__(THREADS)
void rope_kernel(const float* __restrict__ x,
                 const float* __restrict__ sin_c,
                 const float* __restrict__ cos_c,
                 const int*   __restrict__ tpos,
                 float*       __restrict__ out,
                 int S)
{
    // [0 .. 2047]  : sin rows (32 x 64 floats)
    // [2048..4095] : cos rows (32 x 64 floats)
    __shared__ float lds[ROWS_PER_BLOCK * HALF_D * 2];

    const int t  = threadIdx.x;
    const int s0 = blockIdx.x * ROWS_PER_BLOCK;                  // first seq pos
    const size_t R0 = (size_t)blockIdx.y * (size_t)S + (size_t)s0; // first flat row

    // Low 32 bits of a flat shared-aperture address = wave-relative LDS offset.
    const uint32_t lds_base = (uint32_t)(uintptr_t)(void*)lds;

    // ---- Stage sin/cos into LDS: 1024 x 16B chunks, 4 per thread ----
    // Chunk L: L<512 -> sin, else cos; row = (L&511)>>4; 16B segment = L&15.
    // LDS byte offset is exactly L*16 for both halves (cos half starts at 8192).
    #pragma unroll
    for (int j = 0; j < 4; ++j) {
        const int L      = t + j * THREADS;
        const int is_cos = L >> 9;
        const int rr     = (L & 511) >> 4;
        const int seg    = L & 15;
        const int pos    = tpos[s0 + rr];
        const float* src = (is_cos ? cos_c : sin_c) + (size_t)pos * HALF_D + seg * 4;
        const uint32_t ldsa = lds_base + (uint32_t)L * 16u;
        asm volatile("global_load_async_to_lds_b128 %0, %1, off"
                     :
                     : "v"(ldsa), "v"((uint64_t)(uintptr_t)src)
                     : "memory");
    }
    // Drain this wave's async copies, then sync the workgroup so every wave's
    // staged data is visible.
    asm volatile("s_wait_asynccnt 0x0" ::: "memory");
    __syncthreads();

    const float* __restrict__ sin_l = lds;
    const float* __restrict__ cos_l = lds + ROWS_PER_BLOCK * HALF_D;

    // ---- Main stream: 1024 float4s per block, 4 per thread ----
    #pragma unroll
    for (int i = 0; i < 4; ++i) {
        const int f  = i * THREADS + t;   // float4 index within block
        const int r  = f >> 5;            // row 0..31
        const int d4 = f & 31;            // float4 index within row
        const size_t goff = (R0 + (size_t)r) * 128 + (size_t)(d4 * 4);

        const v4f xv = __builtin_nontemporal_load((const v4f*)(x + goff));
        const int k  = d4 * 2;            // pair index
        const v2f sv = *(const v2f*)(sin_l + r * HALF_D + k);
        const v2f cv = *(const v2f*)(cos_l + r * HALF_D + k);

        v4f o;
        o.x = xv.x * cv.x - xv.y * sv.x;  // even' = xe*cos - xo*sin
        o.y = xv.x * sv.x + xv.y * cv.x;  // odd'  = xe*sin + xo*cos
        o.z = xv.z * cv.y - xv.w * sv.y;
        o.w = xv.z * sv.y + xv.w * cv.y;

        __builtin_nontemporal_store(o, (v4f*)(out + goff));
    }
}

extern "C" void kernel_launch(void* const* d_in, const int* in_sizes, int n_in,
                              void* d_out, int out_size, void* d_ws, size_t ws_size,
                              hipStream_t stream) {
    const float* x     = (const float*)d_in[0];
    const float* sin_c = (const float*)d_in[1];
    const float* cos_c = (const float*)d_in[2];
    const int*   tpos  = (const int*)d_in[3];
    float*       out   = (float*)d_out;

    const int S = in_sizes[3];                              // 4096
    const long long rows = (long long)in_sizes[0] / 128;    // B*H*S
    const int sblocks = S / ROWS_PER_BLOCK;                 // 128
    const int bh      = (int)(rows / S);                    // 64

    dim3 grid(sblocks, bh, 1);
    rope_kernel<<<grid, THREADS, 0, stream>>>(x, sin_c, cos_c, tpos, out, S);
}